// ElmanNetwork_58772332478590
// MI455X (gfx1250) — compile-verified
//
#include <hip/hip_runtime.h>
#include <math.h>

typedef float v2f __attribute__((ext_vector_type(2)));
typedef float v8f __attribute__((ext_vector_type(8)));

#define TM 128
#define TN 128
#define TK 32
#define TNP2 144   // lsB2 row stride in float2; 2*144 mod 64 == 32 -> halves hit disjoint banks

// out[m][n] = sum_k X[m][k] * W[n][k]  (+ bias[n]); X:[M,K], W:[N,K] row-major.
// 256 threads (8 waves) per 128x128 tile; wave grid 4x2, each wave 32x64 = 2x4 subtiles
// via V_WMMA_F32_16X16X4_F32. Double-buffered LDS staging; B staged as K-pairs so each
// B fragment is one aligned ds_load_b64 (no repack movs).
template <bool BIAS>
__device__ __forceinline__ void gemm_core(
    const float* __restrict__ X, const float* __restrict__ W,
    const float* __restrict__ bias, float* __restrict__ out,
    int M, int N, int K, int tileM, int tileN, int k0, int k1)
{
    __shared__ float lsA[2][TM * (TK + 2)];      // stride 34: even -> 8B-aligned v2f rows
    __shared__ v2f   lsB2[2][(TK / 2) * TNP2];   // lsB2[k/2][n] = {W[n][k], W[n][k+1]}

    const int t    = threadIdx.x;
    const int lane = t & 31;
    const int w    = t >> 5;
    const int half = lane >> 4;
    const int lr   = lane & 15;
    const int rowbase = (w & 3) * 32;
    const int colbase = (w >> 2) * 64;

    v8f acc[2][4];
#pragma unroll
    for (int im = 0; im < 2; ++im)
#pragma unroll
        for (int jn = 0; jn < 4; ++jn)
            acc[im][jn] = (v8f){0.f, 0.f, 0.f, 0.f, 0.f, 0.f, 0.f, 0.f};

    float4 rA[4], rB[4];

    auto load_regs = [&](int kc) {
#pragma unroll
        for (int i = 0; i < 4; ++i) {
            int idx = t + i * 256;            // float4 units: 128 rows x 8
            int row = idx >> 3, c4 = idx & 7;
            int grow = tileM + row;
            rA[i] = make_float4(0.f, 0.f, 0.f, 0.f);
            if (grow < M)
                rA[i] = *(const float4*)(X + (size_t)grow * K + kc + c4 * 4);
            int gn = tileN + row;             // same tiling for the W tile
            rB[i] = make_float4(0.f, 0.f, 0.f, 0.f);
            if (gn < N)
                rB[i] = *(const float4*)(W + (size_t)gn * K + kc + c4 * 4);
        }
    };

    auto store_lds = [&](int buf) {
#pragma unroll
        for (int i = 0; i < 4; ++i) {
            int idx = t + i * 256;
            int row = idx >> 3, c4 = idx & 7;
            float* dst = &lsA[buf][row * (TK + 2) + c4 * 4];
            dst[0] = rA[i].x; dst[1] = rA[i].y; dst[2] = rA[i].z; dst[3] = rA[i].w;
            lsB2[buf][(c4 * 2 + 0) * TNP2 + row] = (v2f){rB[i].x, rB[i].y};
            lsB2[buf][(c4 * 2 + 1) * TNP2 + row] = (v2f){rB[i].z, rB[i].w};
        }
    };

    const int nchunks = (k1 - k0) / TK;
    load_regs(k0);
    store_lds(0);
    __syncthreads();

    for (int ci = 0; ci < nchunks; ++ci) {
        const int buf = ci & 1;
        if (ci + 1 < nchunks)
            load_regs(k0 + (ci + 1) * TK);    // global loads in flight under the WMMA burst

#pragma unroll
        for (int kk = 0; kk < 8; ++kk) {
            const int k = kk * 4 + 2 * half;  // lanes 16-31 hold K+2/K+3 (ISA layout)
            const int p = kk * 2 + half;      // K-pair row in lsB2
            v2f a[2], b[4];
#pragma unroll
            for (int im = 0; im < 2; ++im) {
                int r = rowbase + im * 16 + lr;
                a[im] = *(const v2f*)&lsA[buf][r * (TK + 2) + k];
            }
#pragma unroll
            for (int jn = 0; jn < 4; ++jn)
                b[jn] = lsB2[buf][p * TNP2 + colbase + jn * 16 + lr];
#pragma unroll
            for (int im = 0; im < 2; ++im)
#pragma unroll
                for (int jn = 0; jn < 4; ++jn)
                    acc[im][jn] = __builtin_amdgcn_wmma_f32_16x16x4_f32(
                        false, a[im], false, b[jn], (short)0, acc[im][jn], false, false);
        }

        if (ci + 1 < nchunks)
            store_lds(buf ^ 1);               // other buffer: no race with this chunk's readers
        __syncthreads();
    }

    // Epilogue: C/D layout -> row = v + 8*half, col = lr
#pragma unroll
    for (int im = 0; im < 2; ++im) {
#pragma unroll
        for (int jn = 0; jn < 4; ++jn) {
            int col = tileN + colbase + jn * 16 + lr;
            float bv = BIAS ? bias[col] : 0.f;
#pragma unroll
            for (int v = 0; v < 8; ++v) {
                int row = tileM + rowbase + im * 16 + v + 8 * half;
                if (row < M)
                    out[(size_t)row * N + col] = acc[im][jn][v] + bv;
            }
        }
    }
}

__global__ __launch_bounds__(256) void k_gemm_splitk(
    const float* __restrict__ X, const float* __restrict__ W, float* __restrict__ part,
    int M, int N, int K, int ksplit)
{
    int s = blockIdx.z;
    gemm_core<false>(X, W, nullptr, part + (size_t)s * M * N,
                     M, N, K, blockIdx.y * TM, blockIdx.x * TN, s * ksplit, (s + 1) * ksplit);
}

__global__ __launch_bounds__(256) void k_gemm_bias(
    const float* __restrict__ X, const float* __restrict__ W,
    const float* __restrict__ bias, float* __restrict__ out, int M, int N, int K)
{
    gemm_core<true>(X, W, bias, out, M, N, K, blockIdx.y * TM, blockIdx.x * TN, 0, K);
}

__global__ __launch_bounds__(256) void k_gemm_nb(
    const float* __restrict__ X, const float* __restrict__ W,
    float* __restrict__ out, int M, int N, int K)
{
    gemm_core<false>(X, W, nullptr, out, M, N, K, blockIdx.y * TM, blockIdx.x * TN, 0, K);
}

// Deterministic split-K reduction (fixed summation order; no fp atomics).
__global__ __launch_bounds__(256) void k_reduce(
    const float* __restrict__ part, float* __restrict__ out, int n, int nsplit, size_t stride)
{
    int i = blockIdx.x * 256 + threadIdx.x;
    if (i < n) {
        float s = 0.f;
        for (int k = 0; k < nsplit; ++k) s += part[(size_t)k * stride + i];
        out[i] = s;
    }
}

// Sequential Elman scan spread over 16 workgroups. Block g owns h rows [32g, 32g+32);
// its W_h slice (32x512 fp32 = 64KB) is LDS-resident for all 511 steps. Per-step
// cross-WGP handoff: 32-float store + device-scope atomic arrive/spin + 512-float read (L2).
__global__ __launch_bounds__(256) void k_elman(
    const float* __restrict__ W_h, const float* __restrict__ xproj,
    float* __restrict__ hs, unsigned* __restrict__ sync, int T1)
{
    const int H = 512, RPB = 32;
    __shared__ float wsl[32 * 513];   // stride 513 -> lane*513 mod 64 distinct: conflict-free
    __shared__ float hbuf[512];
    __shared__ float red[8 * 33];

    const int g = blockIdx.x, t = threadIdx.x;
    const unsigned NB = gridDim.x;

    const float* wsrc = W_h + (size_t)g * RPB * H;
    for (int i = t; i < RPB * H; i += 256) {
        int r = i >> 9, c = i & 511;
        wsl[r * 513 + c] = wsrc[i];
    }
    for (int i = t; i < H; i += 256) hbuf[i] = 0.f;   // h0 = 0
    __syncthreads();

    const int r = t & 31;     // output row within slice (lane == r -> conflict-free wsl reads)
    const int s = t >> 5;     // K-slice (wave-uniform -> hbuf reads are broadcasts)

    for (int step = 0; step < T1; ++step) {
        float acc = 0.f;
        const float* wr = &wsl[r * 513 + s * 64];
        const float* hv = &hbuf[s * 64];
#pragma unroll 8
        for (int k = 0; k < 64; ++k) acc += wr[k] * hv[k];
        red[s * 33 + r] = acc;
        __syncthreads();

        if (t < 32) {
            float sum = 0.f;
#pragma unroll
            for (int ss = 0; ss < 8; ++ss) sum += red[ss * 33 + t];
            float x = xproj[(size_t)step * H + g * RPB + t] + sum;
            float h = 1.f / (1.f + __expf(-x));
            hs[(size_t)step * H + g * RPB + t] = h;
        }
        __syncthreads();

        if (t == 0) {
            __threadfence();                       // release our 32 h values
            atomicAdd(sync, 1u);
            unsigned target = NB * (unsigned)(step + 1);
            while (__hip_atomic_load(sync, __ATOMIC_ACQUIRE, __HIP_MEMORY_SCOPE_AGENT) < target)
                __builtin_amdgcn_s_sleep(1);
        }
        __syncthreads();

        for (int i = t; i < H; i += 256) hbuf[i] = hs[(size_t)step * H + i];
        __syncthreads();
    }
}

// In-place row softmax; logits row (128KB) stays hot in the 192MB L2 across the 3 passes.
__global__ __launch_bounds__(256) void k_softmax(float* __restrict__ data, int cols)
{
    __shared__ float sd[256];
    const int t = threadIdx.x;
    float* p = data + (size_t)blockIdx.x * cols;

    float m = -3.402823466e38f;
    for (int c = t; c < cols; c += 256) m = fmaxf(m, p[c]);
    sd[t] = m; __syncthreads();
    for (int o = 128; o > 0; o >>= 1) { if (t < o) sd[t] = fmaxf(sd[t], sd[t + o]); __syncthreads(); }
    m = sd[0]; __syncthreads();

    float s = 0.f;
    for (int c = t; c < cols; c += 256) { float e = __expf(p[c] - m); p[c] = e; s += e; }
    sd[t] = s; __syncthreads();
    for (int o = 128; o > 0; o >>= 1) { if (t < o) sd[t] += sd[t + o]; __syncthreads(); }
    float inv = 1.f / sd[0];

    for (int c = t; c < cols; c += 256) p[c] *= inv;
}

extern "C" void kernel_launch(void* const* d_in, const int* in_sizes, int n_in,
                              void* d_out, int out_size, void* d_ws, size_t ws_size,
                              hipStream_t stream)
{
    (void)in_sizes; (void)n_in; (void)out_size;
    const float* sent = (const float*)d_in[0];   // [512, 32000]
    const float* W_e  = (const float*)d_in[1];   // [512, 32000]
    const float* W_x  = (const float*)d_in[2];   // [512, 512]
    const float* W_h  = (const float*)d_in[3];   // [512, 512]
    const float* W_p  = (const float*)d_in[4];   // [32000, 512]
    const float* b    = (const float*)d_in[5];   // [512]
    float* out = (float*)d_out;                  // [511, 32000]

    const int T1 = 511, V = 32000, E = 512, H = 512;
    const size_t MN = (size_t)T1 * E;            // 511*512

    // Pick the largest split-K (more blocks -> better occupancy on the V-wide GEMM)
    // whose partial buffer fits in ws and whose k-slice is a multiple of TK=32.
    // ws_size is fixed across calls -> choice is deterministic.
    const int cand[] = {20, 10, 8, 5, 4, 2, 1};  // V/s % 32 == 0 for all of these
    size_t avail = ws_size / sizeof(float);
    int SPLIT = 1;
    for (int i = 0; i < 7; ++i) {
        size_t need = (3 + (size_t)cand[i]) * MN + 64;
        if (need <= avail) { SPLIT = cand[i]; break; }
    }

    float*    emb   = (float*)d_ws;              // MN
    float*    xproj = emb + MN;                  // MN
    float*    hs    = xproj + MN;                // MN
    float*    part  = hs + MN;                   // SPLIT*MN
    unsigned* sync  = (unsigned*)(part + (size_t)SPLIT * MN);

    hipMemsetAsync(sync, 0, 64, stream);         // reset scan barrier each launch

    // emb = sent[:511] @ W_e.T   (split-K over V for occupancy)
    k_gemm_splitk<<<dim3(E / TN, (T1 + TM - 1) / TM, SPLIT), 256, 0, stream>>>(
        sent, W_e, part, T1, E, V, V / SPLIT);
    k_reduce<<<(int)((MN + 255) / 256), 256, 0, stream>>>(part, emb, (int)MN, SPLIT, MN);

    // xproj = emb @ W_x.T + b
    k_gemm_bias<<<dim3(H / TN, (T1 + TM - 1) / TM), 256, 0, stream>>>(
        emb, W_x, b, xproj, T1, H, E);

    // sequential Elman scan (16 cooperating workgroups)
    k_elman<<<16, 256, 0, stream>>>(W_h, xproj, hs, sync, T1);

    // logits = hs @ W_p.T  (250x4 = 1000 blocks)
    k_gemm_nb<<<dim3(V / TN, (T1 + TM - 1) / TM), 256, 0, stream>>>(
        hs, W_p, out, T1, V, H);

    // softmax rows in place
    k_softmax<<<T1, 256, 0, stream>>>(out, V);
}